// DenseGraphWaveletLayer_10316511445514
// MI455X (gfx1250) — compile-verified
//
#include <hip/hip_runtime.h>

#define NN 50000
#define LL 4
#define EE 1600000
#define CC 128

typedef __attribute__((ext_vector_type(2))) float v2f;
typedef __attribute__((ext_vector_type(8))) float v8f;

// ---------------------------------------------------------------------------
// Dense projection: t0[n, c] = sum_k feat[n, scale, k] * W[k, c]
// One wave = one 16x16 output tile via V_WMMA_F32_16X16X4_F32 (32 steps, K=128).
// Block = 256 threads = 8 waves -> 16 rows x 128 cols per block. N = 3125*16.
// ---------------------------------------------------------------------------
__global__ __launch_bounds__(256) void gemm_wmma_f32(
    const float* __restrict__ feat,   // [N, L, C]
    const float* __restrict__ W,      // [C, C] row-major
    float* __restrict__ t0,           // [N, C]
    int scale) {
  __shared__ float As[16][132];       // +4 pad: LDS reads stride-4 banks, conflict-free

  const int tid  = threadIdx.x;
  const int row0 = blockIdx.x * 16;

  // Stage the 16x128 A tile coalesced (8 floats / thread).
#pragma unroll
  for (int it = 0; it < 8; ++it) {
    int idx = it * 256 + tid;
    int r = idx >> 7;
    int c = idx & 127;
    As[r][c] = feat[((size_t)(row0 + r) * LL + scale) * CC + c];
  }
  __syncthreads();

  const int wave = tid >> 5;
  const int lane = tid & 31;
  const int half = lane >> 4;   // ISA: lanes 0-15 hold K=0,1 ; lanes 16-31 hold K=2,3
  const int m    = lane & 15;
  const int n0   = wave * 16;

  v8f acc = {};
#pragma unroll
  for (int kb = 0; kb < CC; kb += 4) {
    v2f a, b;
    // A 16x4 f32 layout: lane m = row m; VGPR0/1 = K = 2*half + {0,1}
    a.x = As[m][kb + 2 * half + 0];
    a.y = As[m][kb + 2 * half + 1];
    // B 4x16 f32 layout (mirrored): lane m = col n0+m; VGPR0/1 = K = 2*half + {0,1}
    b.x = W[(size_t)(kb + 2 * half + 0) * CC + n0 + m];
    b.y = W[(size_t)(kb + 2 * half + 1) * CC + n0 + m];
    acc = __builtin_amdgcn_wmma_f32_16x16x4_f32(
        /*neg_a=*/false, a, /*neg_b=*/false, b,
        /*c_mod=*/(short)0, acc, /*reuse_a=*/false, /*reuse_b=*/false);
  }

  // C/D 16x16 f32 layout: VGPR v, half h -> M = v + 8*h, N = lane%16
#pragma unroll
  for (int v = 0; v < 8; ++v) {
    t0[(size_t)(row0 + v + 8 * half) * CC + n0 + m] = acc[v];
  }
}

// ---------------------------------------------------------------------------
// SpMM pass 1: Y[row] += val * X[col]   (contiguous [N,C] output)
// One wave per nnz; lane owns 4 consecutive channels (float4 gather from
// L2-resident X, 4 non-returning global_atomic_add_f32 scatters).
// ---------------------------------------------------------------------------
__global__ __launch_bounds__(256) void spmm_atomic(
    const int* __restrict__ rows, const int* __restrict__ cols,
    const float* __restrict__ vals,
    const float* __restrict__ X,      // [N, C]
    float* __restrict__ Y) {          // [N, C] (pre-zeroed)
  const int e = (int)((blockIdx.x * 256u + threadIdx.x) >> 5);
  if (e >= EE) return;
  const int lane = threadIdx.x & 31;

  const int   r = rows[e];
  const int   c = cols[e];
  const float v = vals[e];

  const float4 x = *((const float4*)(X + (size_t)c * CC) + lane);
  float* yp = Y + (size_t)r * CC + lane * 4;
  atomicAdd(yp + 0, v * x.x);
  atomicAdd(yp + 1, v * x.y);
  atomicAdd(yp + 2, v * x.z);
  atomicAdd(yp + 3, v * x.w);
}

// ---------------------------------------------------------------------------
// SpMM pass 2 with fused diag(theta):
//   out[row, scale, :] += val * theta[col] * X[col, :]
// Output is the strided [N, L, C] tensor.
// ---------------------------------------------------------------------------
__global__ __launch_bounds__(256) void spmm_atomic_theta_strided(
    const int* __restrict__ rows, const int* __restrict__ cols,
    const float* __restrict__ vals,
    const float* __restrict__ theta,  // [N]
    const float* __restrict__ X,      // [N, C]
    float* __restrict__ out,          // [N, L, C] (pre-zeroed)
    int scale) {
  const int e = (int)((blockIdx.x * 256u + threadIdx.x) >> 5);
  if (e >= EE) return;
  const int lane = threadIdx.x & 31;

  const int   r = rows[e];
  const int   c = cols[e];
  const float v = vals[e] * theta[c];

  const float4 x = *((const float4*)(X + (size_t)c * CC) + lane);
  float* yp = out + ((size_t)r * LL + scale) * CC + lane * 4;
  atomicAdd(yp + 0, v * x.x);
  atomicAdd(yp + 1, v * x.y);
  atomicAdd(yp + 2, v * x.z);
  atomicAdd(yp + 3, v * x.w);
}

// ---------------------------------------------------------------------------
extern "C" void kernel_launch(void* const* d_in, const int* in_sizes, int n_in,
                              void* d_out, int out_size, void* d_ws, size_t ws_size,
                              hipStream_t stream) {
  const int*   phi_idx   = (const int*)d_in[0];    // [L, 2, E]
  const float* phi_val   = (const float*)d_in[1];  // [L, E]
  const int*   phii_idx  = (const int*)d_in[2];    // [L, 2, E]
  const float* phii_val  = (const float*)d_in[3];  // [L, E]
  const float* feat      = (const float*)d_in[4];  // [N, L, C]
  const float* W         = (const float*)d_in[5];  // [C, C]
  const float* theta     = (const float*)d_in[6];  // [N]

  float* out = (float*)d_out;                      // [N, L, C]
  float* t0  = (float*)d_ws;                       // [N, C]
  float* t1  = t0 + (size_t)NN * CC;               // [N, C]

  (void)in_sizes; (void)n_in; (void)out_size; (void)ws_size;

  // Atomic scatter targets must start at zero.
  hipMemsetAsync(out, 0, sizeof(float) * (size_t)NN * LL * CC, stream);

  const int gemm_blocks = NN / 16;                          // 3125
  const int spmm_blocks = (int)(((long)EE * 32 + 255) / 256); // wave per nnz

  for (int i = 0; i < LL; ++i) {
    // 1) y0 = features[:, i, :] @ W   (f32 WMMA)
    gemm_wmma_f32<<<gemm_blocks, 256, 0, stream>>>(feat, W, t0, i);

    // 2) y1 = phi_inverse_i @ y0
    hipMemsetAsync(t1, 0, sizeof(float) * (size_t)NN * CC, stream);
    spmm_atomic<<<spmm_blocks, 256, 0, stream>>>(
        phii_idx + (size_t)(2 * i + 0) * EE,
        phii_idx + (size_t)(2 * i + 1) * EE,
        phii_val + (size_t)i * EE,
        t0, t1);

    // 3+4) out[:, i, :] = phi_i @ (diag(theta) @ y1)   (theta fused into gather)
    spmm_atomic_theta_strided<<<spmm_blocks, 256, 0, stream>>>(
        phi_idx + (size_t)(2 * i + 0) * EE,
        phi_idx + (size_t)(2 * i + 1) * EE,
        phi_val + (size_t)i * EE,
        theta, t1, out, i);
  }
}